// GATLayer_42949673123
// MI455X (gfx1250) — compile-verified
//
#include <hip/hip_runtime.h>
#include <cmath>
#include <cstdint>

typedef __attribute__((ext_vector_type(2))) float v2f;
typedef __attribute__((ext_vector_type(8))) float v8f;

#define DIM 128
#define ROWS 64   // node rows per block in the transform kernel

// ---------------------------------------------------------------------------
// Kernel 0: grid-stride zero fill
// ---------------------------------------------------------------------------
__global__ void gat_zero(float* __restrict__ p, long n) {
    long i = (long)blockIdx.x * blockDim.x + threadIdx.x;
    long stride = (long)gridDim.x * blockDim.x;
    for (; i < n; i += stride) p[i] = 0.0f;
}

// ---------------------------------------------------------------------------
// Kernel 1: fold the first GEMM.
//   s = emb @ W^T + b ; p1 = s·a1 ; p2 = s·a2
//   => p1 = emb·v1 + c1 with v1 = W^T a1, c1 = b·a1 + a_bias (fold a_bias too)
// ---------------------------------------------------------------------------
__global__ void gat_pre(const float* __restrict__ W, const float* __restrict__ b,
                        const float* __restrict__ a, const float* __restrict__ abias,
                        float* __restrict__ v1, float* __restrict__ v2,
                        float* __restrict__ c12) {
    int j = threadIdx.x;  // 0..127
    float s1 = 0.0f, s2 = 0.0f;
    for (int i = 0; i < DIM; ++i) {
        float w = W[i * DIM + j];
        s1 += a[i] * w;
        s2 += a[DIM + i] * w;
    }
    v1[j] = s1;
    v2[j] = s2;
    if (j == 0) {
        float c1 = abias[0], c2 = 0.0f;
        for (int i = 0; i < DIM; ++i) {
            c1 += b[i] * a[i];
            c2 += b[i] * a[DIM + i];
        }
        c12[0] = c1;
        c12[1] = c2;
    }
}

// ---------------------------------------------------------------------------
// Kernel 2: per-node attention projections. One wave32 per node.
// ---------------------------------------------------------------------------
__global__ __launch_bounds__(256) void gat_scores(
    const float* __restrict__ emb, const float* __restrict__ v1,
    const float* __restrict__ v2, const float* __restrict__ c12,
    float* __restrict__ p1, float* __restrict__ p2, int n) {
    int wave = (blockIdx.x * blockDim.x + threadIdx.x) >> 5;
    int lane = threadIdx.x & 31;
    if (wave >= n) return;
    float4 ev = ((const float4*)(emb + (size_t)wave * DIM))[lane];
    float4 w1 = ((const float4*)v1)[lane];
    float4 w2 = ((const float4*)v2)[lane];
    float s1 = ev.x * w1.x + ev.y * w1.y + ev.z * w1.z + ev.w * w1.w;
    float s2 = ev.x * w2.x + ev.y * w2.y + ev.z * w2.z + ev.w * w2.w;
    for (int off = 16; off > 0; off >>= 1) {
        s1 += __shfl_down(s1, off, 32);
        s2 += __shfl_down(s2, off, 32);
    }
    if (lane == 0) {
        p1[wave] = s1 + c12[0];
        p2[wave] = s2 + c12[1];
    }
}

// ---------------------------------------------------------------------------
// Kernel 3: per-edge attention + scatter-add (segment sums via f32 atomics).
// One wave32 per edge; lane handles 4 embedding dims. Gathers and atomics are
// L2-resident (25.6 MB table inside 192 MB L2).
// ---------------------------------------------------------------------------
__global__ __launch_bounds__(256) void gat_edges(
    const int* __restrict__ edges, const float* __restrict__ emb,
    const float* __restrict__ p1, const float* __restrict__ p2,
    float* __restrict__ denom, float* agg, int E) {
    int e = (blockIdx.x * blockDim.x + threadIdx.x) >> 5;
    int lane = threadIdx.x & 31;
    if (e >= E) return;
    int dst = edges[e];
    int src = edges[E + e];
    float score = p1[dst] + p2[src];
    float z = score > 0.0f ? score : 0.2f * score;  // leaky-relu(0.2)
    float attn = __expf(z);
    if (lane == 0) unsafeAtomicAdd(denom + dst, attn);
    float4 ev = ((const float4*)(emb + (size_t)src * DIM))[lane];
    float* ag = agg + (size_t)dst * DIM + lane * 4;
    unsafeAtomicAdd(ag + 0, attn * ev.x);
    unsafeAtomicAdd(ag + 1, attn * ev.y);
    unsafeAtomicAdd(ag + 2, attn * ev.z);
    unsafeAtomicAdd(ag + 3, attn * ev.w);
}

// ---------------------------------------------------------------------------
// Kernel 4: h = [emb | agg/denom] @ transW^T + tb, then LayerNorm.
//
// Block = 256 threads = 8 waves; each block owns ROWS=64 nodes. Wave w owns
// output columns [16w,16w+16) for all 4 row tiles: one B fragment feeds FOUR
// v_wmma_f32_16x16x4_f32 ops (4 accumulators), quartering trans_weight L2
// traffic vs a 16-row tile. K = 256 -> 64 iters x 4 WMMAs per wave.
//
// The emb half of the A tile is staged with global_load_async_to_lds_b128
// (ASYNCcnt path); the agg half needs a VALU scale so it uses regular loads,
// overlapping with the async copies, then s_wait_asynccnt 0 + barrier.
//
// f32 16x16x4 fragment layouts (ISA 7.12.2):
//   A: lane L<16 -> (M=L, K=0..1); lane L>=16 -> (M=L-16, K=2..3)
//   B: lanes 0-15 rows K=0..1 at N=lane; lanes 16-31 rows K=2..3
//   C/D: lane<16 VGPR r -> (M=r, N=lane); lane>=16 VGPR r -> (M=r+8, N=lane-16)
// ---------------------------------------------------------------------------
__global__ __launch_bounds__(256) void gat_trans(
    const float* __restrict__ emb, const float* agg,
    const float* __restrict__ denom, const float* __restrict__ tW,
    const float* __restrict__ tb, const float* __restrict__ gamma,
    const float* __restrict__ beta, float* out, int n) {
    __shared__ float ldsA[ROWS][2 * DIM];   // 64 KB: concat([emb, agg_norm])
    __shared__ float ldsH[ROWS][DIM];       // 32 KB: pre-LN output tile
    __shared__ float rden[ROWS];
    __shared__ float mu_s[ROWS];
    __shared__ float rs_s[ROWS];

    int tid = threadIdx.x;
    int row0 = blockIdx.x * ROWS;

    if (tid < ROWS) {
        int node = row0 + tid;
        float d = (node < n) ? denom[node] : 1.0f;
        rden[tid] = 1.0f / (d + 1e-20f);
    }

    // ---- Stage emb half via async global->LDS copies (16 B per op). ----
    // ROWS * (DIM/4) = 2048 float4 chunks, 8 per thread.
    for (int idx = tid; idx < ROWS * (DIM / 4); idx += 256) {
        int r  = idx >> 5;             // DIM/4 = 32 chunks per row
        int c4 = (idx & 31) * 4;
        int node = row0 + r;
        if (node < n) {
            unsigned lds_off = (unsigned)(uintptr_t)(&ldsA[r][c4]);
            const float* gptr = emb + (size_t)node * DIM + c4;
            asm volatile("global_load_async_to_lds_b128 %0, %1, off"
                         :: "v"(lds_off), "v"(gptr)
                         : "memory");
        } else {
            *(float4*)&ldsA[r][c4] = make_float4(0.f, 0.f, 0.f, 0.f);
        }
    }
    __syncthreads();  // rden visible before the agg staging below

    // ---- Stage agg half (needs 1/denom scale) with regular loads. ----
    for (int idx = tid; idx < ROWS * (DIM / 4); idx += 256) {
        int r  = idx >> 5;
        int c4 = (idx & 31) * 4;
        int node = row0 + r;
        float4 v = make_float4(0.f, 0.f, 0.f, 0.f);
        if (node < n) {
            float4 t = *(const float4*)(agg + (size_t)node * DIM + c4);
            float s = rden[r];
            v = make_float4(t.x * s, t.y * s, t.z * s, t.w * s);
        }
        *(float4*)&ldsA[r][DIM + c4] = v;
    }
    asm volatile("s_wait_asynccnt 0x0" ::: "memory");  // my async copies landed
    __syncthreads();                                   // everyone's landed

    // ---- WMMA accumulation: 4 row tiles per wave, shared B fragment. ----
    int wave = tid >> 5;           // 0..7 -> output column tile
    int lane = tid & 31;
    int colBase = wave * 16;
    int m = lane & 15;             // row (A) / col (B) index within tile
    int khalf = (lane >> 4) * 2;   // 0 for lanes 0-15, 2 for lanes 16-31

    v8f acc0 = {}, acc1 = {}, acc2 = {}, acc3 = {};
    const float* brow = tW + (size_t)(colBase + m) * (2 * DIM);  // trans_weight[o,:]
    for (int kk = 0; kk < 2 * DIM; kk += 4) {
        v2f bf;
        bf.x = brow[kk + khalf + 0];
        bf.y = brow[kk + khalf + 1];
        v2f a0, a1, a2, a3;
        a0.x = ldsA[m +  0][kk + khalf]; a0.y = ldsA[m +  0][kk + khalf + 1];
        a1.x = ldsA[m + 16][kk + khalf]; a1.y = ldsA[m + 16][kk + khalf + 1];
        a2.x = ldsA[m + 32][kk + khalf]; a2.y = ldsA[m + 32][kk + khalf + 1];
        a3.x = ldsA[m + 48][kk + khalf]; a3.y = ldsA[m + 48][kk + khalf + 1];
        acc0 = __builtin_amdgcn_wmma_f32_16x16x4_f32(false, a0, false, bf,
                                                     (short)0, acc0, false, false);
        acc1 = __builtin_amdgcn_wmma_f32_16x16x4_f32(false, a1, false, bf,
                                                     (short)0, acc1, false, false);
        acc2 = __builtin_amdgcn_wmma_f32_16x16x4_f32(false, a2, false, bf,
                                                     (short)0, acc2, false, false);
        acc3 = __builtin_amdgcn_wmma_f32_16x16x4_f32(false, a3, false, bf,
                                                     (short)0, acc3, false, false);
    }

    // ---- Spill accumulator tiles (+bias) to LDS per the f32 C/D layout. ----
    int nIdx = lane & 15;
    int rowAdd = (lane >> 4) * 8;
    float tbv = tb[colBase + nIdx];
#pragma unroll
    for (int r = 0; r < 8; ++r) {
        ldsH[ 0 + rowAdd + r][colBase + nIdx] = acc0[r] + tbv;
        ldsH[16 + rowAdd + r][colBase + nIdx] = acc1[r] + tbv;
        ldsH[32 + rowAdd + r][colBase + nIdx] = acc2[r] + tbv;
        ldsH[48 + rowAdd + r][colBase + nIdx] = acc3[r] + tbv;
    }
    __syncthreads();

    // ---- LayerNorm statistics: one thread per row. ----
    if (tid < ROWS) {
        float s = 0.0f;
        for (int c = 0; c < DIM; ++c) s += ldsH[tid][c];
        float mu = s * (1.0f / DIM);
        float v = 0.0f;
        for (int c = 0; c < DIM; ++c) {
            float d = ldsH[tid][c] - mu;
            v += d * d;
        }
        mu_s[tid] = mu;
        rs_s[tid] = rsqrtf(v * (1.0f / DIM) + 1e-5f);
    }
    __syncthreads();

    // ---- Normalize + affine + writeback (overwrites this block's agg rows). ----
    for (int idx = tid; idx < ROWS * DIM; idx += 256) {
        int r = idx >> 7;
        int c = idx & 127;
        int node = row0 + r;
        if (node < n)
            out[(size_t)node * DIM + c] =
                (ldsH[r][c] - mu_s[r]) * rs_s[r] * gamma[c] + beta[c];
    }
}

// ---------------------------------------------------------------------------
// Launcher
// ---------------------------------------------------------------------------
extern "C" void kernel_launch(void* const* d_in, const int* in_sizes, int n_in,
                              void* d_out, int out_size, void* d_ws, size_t ws_size,
                              hipStream_t stream) {
    const float* emb   = (const float*)d_in[0];
    const int*   edges = (const int*)d_in[1];
    const float* W     = (const float*)d_in[2];
    const float* b     = (const float*)d_in[3];
    const float* a     = (const float*)d_in[4];
    const float* abias = (const float*)d_in[5];
    const float* tW    = (const float*)d_in[6];
    const float* tb    = (const float*)d_in[7];
    const float* gamma = (const float*)d_in[8];
    const float* beta  = (const float*)d_in[9];

    int N = in_sizes[0] / DIM;   // 50000
    int E = in_sizes[1] / 2;     // 800000

    float* out = (float*)d_out;
    float* ws  = (float*)d_ws;

    // Workspace (~600 KB): v1|v2|c|p1|p2|denom. agg aliases d_out: it is
    // zeroed every launch and each trans block reads its agg rows into LDS
    // before overwriting them with the final output.
    int NP = (N + 127) & ~127;
    float* v1    = ws;
    float* v2    = ws + DIM;
    float* c12   = ws + 2 * DIM;
    float* p1    = ws + 4 * DIM;
    float* p2    = p1 + NP;
    float* denom = p2 + NP;
    float* agg   = out;

    gat_zero<<<2048, 256, 0, stream>>>(agg, (long)N * DIM);
    gat_zero<<<64, 256, 0, stream>>>(denom, (long)N);
    gat_pre<<<1, DIM, 0, stream>>>(W, b, a, abias, v1, v2, c12);

    int scoreBlocks = (int)(((long)N * 32 + 255) / 256);
    gat_scores<<<scoreBlocks, 256, 0, stream>>>(emb, v1, v2, c12, p1, p2, N);

    int edgeBlocks = (int)(((long)E * 32 + 255) / 256);
    gat_edges<<<edgeBlocks, 256, 0, stream>>>(edges, emb, p1, p2, denom, agg, E);

    gat_trans<<<(N + ROWS - 1) / ROWS, 256, 0, stream>>>(emb, agg, denom, tW, tb,
                                                         gamma, beta, out, N);
}